// ForwardAttributeGnn_43284680409240
// MI455X (gfx1250) — compile-verified
//
#include <hip/hip_runtime.h>
#include <stddef.h>
#include <stdint.h>

// ---------------------------------------------------------------------------
// MI455X (gfx1250) fused Attribute-GNN forward.
//   out[b,a,:] = concat(src[b], edge[b,a]) @ QA[a] + c[a]
//   QA[a] = W_agg^T @ proj[a]  (precomputed, 1MB -> L2 resident)
//   c[a]  = b_agg   @ proj[a]
// Main kernel: wave32 V_WMMA_F32_16X16X4_F32, one wave = 16 batch rows.
// QA stored K-transposed so each B-fragment is one contiguous (ds) b64.
// LDS: ping-pong double buffer of 64-j half tiles filled with
// GLOBAL_LOAD_ASYNC_TO_LDS_B128 (ASYNCcnt-tracked) + LDS-resident bias table,
// eliminating the per-tile global bias stall and the staging bubble.
// ---------------------------------------------------------------------------

typedef __attribute__((ext_vector_type(2))) float v2f;
typedef __attribute__((ext_vector_type(4))) float v4f;
typedef __attribute__((ext_vector_type(8))) float v8f;
typedef __attribute__((ext_vector_type(4))) int   v4i;

typedef __attribute__((address_space(1))) v4i g_v4i;   // global
typedef __attribute__((address_space(3))) v4i l_v4i;   // LDS

#define A_TYPES 8
#define EDIM    128
#define K2      256
#define PITCH   260                 // floats; 260 % 64 == 4 -> conflict-free b64
#define HALF_J  64                  // j-rows per LDS buffer (half a QA tile)
#define BUF_WORDS (HALF_J * PITCH)  // 16640 floats = 66,560 B per buffer

#if __has_builtin(__builtin_amdgcn_global_load_async_to_lds_b128)
#define USE_ASYNC_LDS 1
#else
#define USE_ASYNC_LDS 0
#endif

__device__ __forceinline__ void wait_async_zero() {
#if USE_ASYNC_LDS
#if __has_builtin(__builtin_amdgcn_s_wait_asynccnt)
    __builtin_amdgcn_s_wait_asynccnt(0);
#else
    asm volatile("s_wait_asynccnt 0x0" ::: "memory");
#endif
#endif
}

// Copy one 64x256 f32 half-tile of QA_T[a] into a padded LDS buffer.
__device__ __forceinline__ void stage_half(const float* __restrict__ qaT,
                                           int stageIdx, float* dst, int tid) {
    const int a    = stageIdx >> 1;
    const int half = stageIdx & 1;
    const float* g = qaT + (size_t)(a * EDIM + half * HALF_J) * K2;
#if USE_ASYNC_LDS
    #pragma unroll
    for (int it = 0; it < 16; ++it) {
        const int idx = tid + it * 256;          // 0..4095 float4s
        const int row = idx >> 6;                // 64 float4 per 256-float row
        const int q   = idx & 63;
        __builtin_amdgcn_global_load_async_to_lds_b128(
            (g_v4i*)(g + (size_t)idx * 4),
            (l_v4i*)(dst + row * PITCH + q * 4),
            0, 0);
    }
#else
    const v4f* g4 = (const v4f*)g;
    #pragma unroll
    for (int it = 0; it < 16; ++it) {
        const int idx = tid + it * 256;
        const int row = idx >> 6;
        const int q   = idx & 63;
        *(v4f*)(dst + row * PITCH + q * 4) = g4[idx];
    }
#endif
}

// ---------------------------------------------------------------------------
// Precompute QA_T[a][j][k] = sum_i W_agg[i][k] * proj[a][i][j]
//            cbias[a][j]   = sum_i b_agg[i]    * proj[a][i][j]
// ---------------------------------------------------------------------------
__global__ __launch_bounds__(256) void qa_precompute_kernel(
    const float* __restrict__ Wagg, const float* __restrict__ bagg,
    const float* __restrict__ proj, float* __restrict__ qaT,
    float* __restrict__ cbias)
{
    const int blk = blockIdx.x;          // 0 .. A*128-1
    const int a   = blk >> 7;
    const int j   = blk & 127;
    const int k   = threadIdx.x;         // 0..255

    const float* pa = proj + ((size_t)a * EDIM) * EDIM + j;
    const float* w  = Wagg + k;

    float acc = 0.0f, cacc = 0.0f;
    #pragma unroll 8
    for (int i = 0; i < EDIM; ++i) {
        const float p = pa[(size_t)i * EDIM];   // uniform per block
        acc  += w[(size_t)i * K2] * p;          // coalesced over k
        cacc += bagg[i] * p;
    }
    qaT[((size_t)a * EDIM + j) * K2 + k] = acc;
    if (k == 0) cbias[a * EDIM + j] = cacc;
}

// ---------------------------------------------------------------------------
// Main fused kernel: 256 threads = 8 waves; wave w owns 16 batch rows.
// ---------------------------------------------------------------------------
__global__ __launch_bounds__(256) void gnn_fused_wmma_kernel(
    const float* __restrict__ src,    // [B][EDIM]
    const float* __restrict__ edge,   // [B][A][EDIM]
    const float* __restrict__ qaT,    // [A][EDIM(j)][K2(k)]
    const float* __restrict__ cbias,  // [A][EDIM]
    float* __restrict__ out)          // [B][A][EDIM]
{
    __shared__ __align__(16) float qa[2][BUF_WORDS];     // 133,120 B
    __shared__ __align__(16) float cbl[A_TYPES * EDIM];  //   4,096 B

    const int tid  = threadIdx.x;
    const int lane = tid & 31;
    const int wave = tid >> 5;
    const int ln   = lane & 15;          // M-row (A frag) / N-col (B,C frags)
    const int hi   = lane >> 4;          // 0 | 1
    const int koff = hi * 2;             // K sub-offset for upper half-wave
    const int base_b = blockIdx.x * 128 + wave * 16;

    // ---- bias table -> LDS (one b128 per thread)
    *(v4f*)(cbl + tid * 4) = *(const v4f*)(cbias + tid * 4);

    // ---- src A-fragments: register resident, reused for all a and tiles
    v2f sa[32];
    {
        const float* sp = src + (size_t)(base_b + ln) * EDIM + koff;
        #pragma unroll
        for (int kb = 0; kb < 32; ++kb)
            sa[kb] = *(const v2f*)(sp + kb * 4);
    }

    // ---- prologue: stage first half-tile
    stage_half(qaT, 0, qa[0], tid);
    wait_async_zero();
    __syncthreads();

    v2f ea[32];
    for (int st = 0; st < 2 * A_TYPES; ++st) {           // 16 stages
        const int buf  = st & 1;
        const int a    = st >> 1;
        const int half = st & 1;

        // stage next half into the other buffer (overlaps with compute below;
        // that buffer was last read in stage st-1, fenced by the last barrier)
        if (st < 2 * A_TYPES - 1)
            stage_half(qaT, st + 1, qa[buf ^ 1], tid);

        if (half == 0) {                                  // new a: edge frags
            const float* ep =
                edge + ((size_t)(base_b + ln) * A_TYPES + a) * EDIM + koff;
            #pragma unroll
            for (int kb = 0; kb < 32; ++kb)
                ea[kb] = *(const v2f*)(ep + kb * 4);
            if (a < A_TYPES - 1)
                __builtin_prefetch(ep + EDIM, 0, 3);      // warm next a
        }

        const float* qb = qa[buf];
        for (int tt = 0; tt < 4; ++tt) {                  // 4 j-tiles per half
            const int t = half * 4 + tt;
            const float bias = cbl[a * EDIM + t * 16 + ln];  // ds_load, cheap
            v8f c;
            #pragma unroll
            for (int r = 0; r < 8; ++r) c[r] = bias;

            const float* bbase = qb + (tt * 16 + ln) * PITCH + koff;

            #pragma unroll
            for (int kb = 0; kb < 32; ++kb) {             // K = 0..127 (src)
                const v2f b = *(const v2f*)(bbase + kb * 4);
                c = __builtin_amdgcn_wmma_f32_16x16x4_f32(
                        false, sa[kb], false, b, (short)0, c, false, false);
            }
            #pragma unroll
            for (int kb = 0; kb < 32; ++kb) {             // K = 128..255 (edge)
                const v2f b = *(const v2f*)(bbase + EDIM + kb * 4);
                c = __builtin_amdgcn_wmma_f32_16x16x4_f32(
                        false, ea[kb], false, b, (short)0, c, false, false);
            }

            // D vgpr r, this lane -> (m = r + 8*hi, j = t*16 + ln)
            float* op = out + ((size_t)(base_b + hi * 8) * A_TYPES + a) * EDIM
                            + t * 16 + ln;
            #pragma unroll
            for (int r = 0; r < 8; ++r)
                op[(size_t)r * A_TYPES * EDIM] = c[r];
        }

        wait_async_zero();     // our async stage of st+1 landed in LDS
        __syncthreads();       // everyone's did; buffers safe to swap
    }
}

// ---------------------------------------------------------------------------
// Launch. Inputs: 0=image_features [B,E], 1=edge_attributes [B,A,E],
// 2=W_agg [E,2E], 3=b_agg [E], 4=proj [A,E,E]. Workspace: QA_T (1MB) + c (4KB).
// ---------------------------------------------------------------------------
extern "C" void kernel_launch(void* const* d_in, const int* in_sizes, int n_in,
                              void* d_out, int out_size, void* d_ws, size_t ws_size,
                              hipStream_t stream) {
    const float* img  = (const float*)d_in[0];
    const float* edge = (const float*)d_in[1];
    const float* Wagg = (const float*)d_in[2];
    const float* bagg = (const float*)d_in[3];
    const float* proj = (const float*)d_in[4];
    float* out = (float*)d_out;

    float* qaT = (float*)d_ws;                          // A*128*256 floats
    float* cb  = qaT + (size_t)A_TYPES * EDIM * K2;     // A*128 floats

    const int B = in_sizes[0] / EDIM;                   // 65536

    qa_precompute_kernel<<<dim3(A_TYPES * EDIM), dim3(256), 0, stream>>>(
        Wagg, bagg, proj, qaT, cb);

    gnn_fused_wmma_kernel<<<dim3(B / 128), dim3(256), 0, stream>>>(
        img, edge, qaT, cb, out);
}